// DenoisingMedianNonDifferentiable_52785148068078
// MI455X (gfx1250) — compile-verified
//
#include <hip/hip_runtime.h>
#include <stdint.h>

// ---------------- problem constants ----------------
#define IMG_W 512
#define IMG_H 512
#define HW (IMG_W * IMG_H)       // 262144 = 2^18
#define NB 4
#define NPIX (NB * HW)           // 1048576 total pixels (all batches)

// median kernel tiling
#define TW 32
#define TH 8
#define CP 4                     // chroma pad (9x9 window)
#define CTW (TW + 2 * CP)        // 40
#define CTH (TH + 2 * CP)        // 16
#define YTW (TW + 2)             // 34
#define YTH (TH + 2)             // 10

typedef __attribute__((ext_vector_type(2))) float v2f;
typedef __attribute__((ext_vector_type(8))) float v8f;

#if __has_builtin(__builtin_amdgcn_global_load_async_to_lds_b32)
#define USE_ASYNC_LDS 1
typedef __attribute__((address_space(1))) int as1_int;
typedef __attribute__((address_space(3))) int as3_int;
#else
#define USE_ASYNC_LDS 0
#endif

// ---------------- helpers ----------------
__device__ __forceinline__ void cex(float& a, float& b) {
  float lo = fminf(a, b);
  float hi = fmaxf(a, b);
  a = lo;
  b = hi;
}

__device__ __forceinline__ float med3(float a, float b, float c) {
#if __has_builtin(__builtin_amdgcn_fmed3f)
  return __builtin_amdgcn_fmed3f(a, b, c);   // v_med3_f32
#else
  return fmaxf(fminf(a, b), fminf(fmaxf(a, b), c));
#endif
}

// numpy 'symmetric' (= scipy 'reflect') index mapping; single reflection
// suffices since pad (<=4) << 512.
__device__ __forceinline__ int reflect_idx(int i, int n) {
  i = (i < 0) ? (-1 - i) : i;
  i = (i >= n) ? (2 * n - 1 - i) : i;
  return i;
}

// Classic 19-exchange median-of-9 network (Paeth).
__device__ __forceinline__ float median9(float p0, float p1, float p2,
                                         float p3, float p4, float p5,
                                         float p6, float p7, float p8) {
  cex(p1, p2); cex(p4, p5); cex(p7, p8);
  cex(p0, p1); cex(p3, p4); cex(p6, p7);
  cex(p1, p2); cex(p4, p5); cex(p7, p8);
  cex(p0, p3); cex(p5, p8); cex(p4, p7);
  cex(p3, p6); cex(p1, p4); cex(p2, p5);
  cex(p4, p7); cex(p4, p2); cex(p6, p4);
  cex(p4, p2);
  return p4;
}

// Exact median of the 81 elements sm[(i/9)*stride + i%9], i = 0..80,
// via forgetful selection with a 42-element working set kept in VGPRs.
// Min/max of each pass are found with a pairwise compare-exchange pass
// (disjoint pairs, then a min chain over pair-min slots and a max chain
// over pair-max slots): ~1.5n ops instead of 2n, multiset preserved at
// every step so the selection argument is untouched.
__device__ __forceinline__ float median81(const float* sm, const int stride) {
  float w[42];
#pragma unroll
  for (int i = 0; i < 42; ++i) w[i] = sm[(i / 9) * stride + (i % 9)];

  // Phase A: 20 eliminate-min&max passes over the fixed range [0,41] (n=42,
  // even), streaming the 39 unseen window elements in as replacements.
#pragma unroll
  for (int t = 0; t < 20; ++t) {
#pragma unroll
    for (int j = 0; j < 21; ++j) cex(w[2 * j], w[2 * j + 1]);   // pairs
#pragma unroll
    for (int j = 1; j < 21; ++j) cex(w[0], w[2 * j]);           // min -> w[0]
#pragma unroll
    for (int j = 0; j < 20; ++j) cex(w[2 * j + 1], w[41]);      // max -> w[41]
    if (t < 19) {
      const int e0 = 42 + 2 * t;
      const int e1 = 43 + 2 * t;
      w[0]  = sm[(e0 / 9) * stride + (e0 % 9)];
      w[41] = sm[(e1 / 9) * stride + (e1 % 9)];
    } else {
      w[0] = sm[(80 / 9) * stride + (80 % 9)];
    }
  }

  // Phase B: 41 candidates in w[0..40] (20 too-small and 20 too-large were
  // discarded, so their median is the answer). Peel min/max pairs: range
  // [lo,hi] has odd n = 41-2t; leftover element w[hi] is folded into both
  // chains by an initial cex(w[lo], w[hi]).
#pragma unroll
  for (int t = 0; t < 19; ++t) {
    const int lo = t, hi = 40 - t;
    const int half = 20 - t;                                    // (n-1)/2
#pragma unroll
    for (int j = 0; j < half; ++j) cex(w[lo + 2 * j], w[lo + 2 * j + 1]);
    cex(w[lo], w[hi]);
#pragma unroll
    for (int j = 1; j < half; ++j) cex(w[lo], w[lo + 2 * j]);   // min -> w[lo]
#pragma unroll
    for (int j = 0; j < half; ++j) cex(w[lo + 2 * j + 1], w[hi]); // max -> w[hi]
  }
  return med3(w[19], w[20], w[21]);
}

// ---------------- kernel 1: RGB -> YUV via V_WMMA_F32_16X16X4_F32 ----------
// D(16x16) = A(16x4) * B(4x16): A holds the 3x3 RGB2YUV matrix (rows 0..2,
// cols 0..2; rest zero), B holds 16 pixels' RGB (rows K=0..2), so D rows
// 0..2 are the Y/U/V of 16 pixels. Per wave: load all 4 B fragments as one
// clause, run 4 back-to-back WMMAs, then store -- keeps the XDL pipe fed.
__global__ __launch_bounds__(256) void k_rgb2yuv(const float* __restrict__ img,
                                                 float* __restrict__ yuv) {
  const int lane = threadIdx.x & 31;
  const int wid  = threadIdx.x >> 5;
  const int m    = lane & 15;       // M row (A) / N col (B,D)
  const bool hi  = lane >= 16;      // lane half selects K=0/1 vs K=2/3

  // A fragment: VGPR0 = col (hi?2:0) of row m, VGPR1 = col (hi?3:1).
  float c0 = (m == 0) ? 0.299f : (m == 1) ? -0.14713f : (m == 2) ? 0.615f    : 0.0f;
  float c1 = (m == 0) ? 0.587f : (m == 1) ? -0.28886f : (m == 2) ? -0.51499f : 0.0f;
  float c2 = (m == 0) ? 0.114f : (m == 1) ? 0.436f    : (m == 2) ? -0.10001f : 0.0f;
  v2f A;
  A.x = hi ? c2 : c0;
  A.y = hi ? 0.0f : c1;   // col 3 is zero -> B's K=3 row is don't-care

  float* Yp = yuv;
  float* Up = yuv + NPIX;
  float* Vp = yuv + 2 * NPIX;

  const int gbase = (blockIdx.x * 8 + wid) * 64;

  v2f Bv[4];
#pragma unroll
  for (int u = 0; u < 4; ++u) {
    const int g = gbase + u * 16 + m;        // pixel id (group never splits a batch)
    const int b = g >> 18;                   // HW == 2^18
    const int q = g & (HW - 1);
    const float* plane = img + (size_t)b * (3 * HW);
    Bv[u].x = plane[(hi ? 2 * HW : 0) + q];  // K=0 (ch0) / K=2 (ch2)
    Bv[u].y = plane[HW + q];                 // K=1 (ch1) / K=3 (don't care)
  }

  v8f D[4];
#pragma unroll
  for (int u = 0; u < 4; ++u) {
    v8f C = {};
    D[u] = __builtin_amdgcn_wmma_f32_16x16x4_f32(false, A, false, Bv[u],
                                                 (short)0, C, false, false);
  }

  if (!hi) {                                 // D rows 0..2 live in lanes 0..15
#pragma unroll
    for (int u = 0; u < 4; ++u) {
      const int g = gbase + u * 16 + m;
      Yp[g] = D[u][0];
      Up[g] = D[u][1];
      Vp[g] = D[u][2];
    }
  }
}

// ------- kernel 2: LDS-tiled medians + fused YUV -> RGB ---------------------
__global__ __launch_bounds__(256) void k_median_yuv2rgb(const float* __restrict__ yuv,
                                                        float* __restrict__ out) {
  __shared__ float sY[YTH][YTW];
  __shared__ float sU[CTH][CTW];
  __shared__ float sV[CTH][CTW];

  const int b   = blockIdx.z;
  const int x0  = blockIdx.x * TW;
  const int y0  = blockIdx.y * TH;
  const int tid = threadIdx.y * TW + threadIdx.x;

  const float* Yp = yuv + (size_t)b * HW;
  const float* Up = yuv + NPIX + (size_t)b * HW;
  const float* Vp = yuv + 2 * NPIX + (size_t)b * HW;

  // chroma tiles (symmetric halo of 4) via async global->LDS (ASYNCcnt path)
  for (int i = tid; i < CTH * CTW; i += 256) {
    const int ty = i / CTW, tx = i % CTW;
    const int gy = reflect_idx(y0 - CP + ty, IMG_H);
    const int gx = reflect_idx(x0 - CP + tx, IMG_W);
    const int gi = gy * IMG_W + gx;
#if USE_ASYNC_LDS
    __builtin_amdgcn_global_load_async_to_lds_b32(
        (as1_int*)(Up + gi), (as3_int*)&sU[ty][tx], 0, 0);
    __builtin_amdgcn_global_load_async_to_lds_b32(
        (as1_int*)(Vp + gi), (as3_int*)&sV[ty][tx], 0, 0);
#else
    sU[ty][tx] = Up[gi];
    sV[ty][tx] = Vp[gi];
#endif
  }
  // luma tile (halo of 1)
  for (int i = tid; i < YTH * YTW; i += 256) {
    const int ty = i / YTW, tx = i % YTW;
    const int gy = reflect_idx(y0 - 1 + ty, IMG_H);
    const int gx = reflect_idx(x0 - 1 + tx, IMG_W);
    sY[ty][tx] = Yp[gy * IMG_W + gx];
  }
#if USE_ASYNC_LDS
  asm volatile("s_wait_asynccnt 0" ::: "memory");
#endif
  __syncthreads();

  const int lx = threadIdx.x, ly = threadIdx.y;

  float ym = median9(sY[ly + 0][lx + 0], sY[ly + 0][lx + 1], sY[ly + 0][lx + 2],
                     sY[ly + 1][lx + 0], sY[ly + 1][lx + 1], sY[ly + 1][lx + 2],
                     sY[ly + 2][lx + 0], sY[ly + 2][lx + 1], sY[ly + 2][lx + 2]);
  ym = fminf(fmaxf(ym, 0.0f), 1.0f);

  const float um = median81(&sU[ly][lx], CTW);
  const float vm = median81(&sV[ly][lx], CTW);

  // YUV2RGB = [[1,0,1.13983],[1,-0.39465,-0.5806],[1,2.03211,0]]
  const float r  = ym + 1.13983f * vm;
  const float g  = ym - 0.39465f * um - 0.5806f * vm;
  const float bl = ym + 2.03211f * um;

  const size_t o = (size_t)(b * 3) * HW + (size_t)(y0 + ly) * IMG_W + (x0 + lx);
  out[o]          = r;
  out[o + HW]     = g;
  out[o + 2 * HW] = bl;
}

// ---------------- launch ----------------
extern "C" void kernel_launch(void* const* d_in, const int* in_sizes, int n_in,
                              void* d_out, int out_size, void* d_ws, size_t ws_size,
                              hipStream_t stream) {
  (void)in_sizes; (void)n_in; (void)out_size; (void)ws_size;
  const float* img = (const float*)d_in[0];
  float* out = (float*)d_out;
  float* yuv = (float*)d_ws;  // needs 3 * NPIX * 4 = 12.6 MB scratch

  k_rgb2yuv<<<NPIX / 512, 256, 0, stream>>>(img, yuv);

  dim3 grid(IMG_W / TW, IMG_H / TH, NB);
  dim3 block(TW, TH);
  k_median_yuv2rgb<<<grid, block, 0, stream>>>(yuv, out);
}